// MaskingMixture_73289321939081
// MI455X (gfx1250) — compile-verified
//
#include <hip/hip_runtime.h>
#include <stdint.h>

// MaskingMixture top-K selection for MI455X (gfx1250).
// Bandwidth-bound top-K/rank workload: no GEMM structure -> WMMA inapplicable.
// CDNA5 path: double-buffered async global->LDS staging
// (global_load_async_to_lds_b128 + in-order ASYNCcnt, s_wait_asynccnt N)
// feeding LDS histogram radix-select passes. Keys (64 MB) are L2-resident
// (192 MB L2), so passes 2-4 stream from L2, pass 1 from HBM.

#define BATCH 128
#define NTOK  131072          // T*H*W = 64*64*32
#define TDIM  64
#define HWREP 2048            // H*W
#define SEL_THREADS 256
#define TILE_U32 4096         // 16 KB staging tile = 256 threads * 64 B
#define TILE_BYTES (TILE_U32 * 4)
#define NTILES (NTOK / TILE_U32)   // 32 tiles per row pass

typedef unsigned int uint32;

__device__ __forceinline__ uint32 f2key(float f) {
  uint32 u = __float_as_uint(f);
  return (u & 0x80000000u) ? ~u : (u | 0x80000000u);  // order-preserving
}

// raw v_log_f32: inputs are in [1e-3, 1) -- no denormal fixup needed
__device__ __forceinline__ float fastlog(float x) {
  return __builtin_amdgcn_logf(x) * 0.6931471805599453f;  // log2 -> ln
}

// ---------------- key construction ----------------
__global__ void k_keys_src(const float* __restrict__ U0,
                           const float* __restrict__ Ut,
                           uint32* __restrict__ keys) {
  size_t i4 = ((size_t)blockIdx.x * blockDim.x + threadIdx.x) * 4;
  int b = (int)(i4 / NTOK);
  int j = (int)(i4 % NTOK);
  const float4 u0 = *(const float4*)(U0 + i4);
  float uv[4] = {u0.x, u0.y, u0.z, u0.w};
  uint32 res[4];
#pragma unroll
  for (int q = 0; q < 4; ++q) {
    int t = (j + q) / HWREP;
    // linspace(1, 0.001, 64)[t] ;  log(L^(1/3)) = log(L)/3
    float lin = 1.0f + (float)t * ((0.001f - 1.0f) / 63.0f);
    float P = fastlog(uv[q]) + 0.5f * fastlog(Ut[b * TDIM + t])
            + (1.0f / 3.0f) * fastlog(lin);
    res[q] = f2key(P);
  }
  *(uint4*)(keys + i4) = make_uint4(res[0], res[1], res[2], res[3]);
}

__global__ void k_keys_tgt(const float* __restrict__ U0,
                           const float* __restrict__ Ut,
                           const float* __restrict__ srcmask,  // 0/1 floats
                           uint32* __restrict__ keys) {
  size_t i4 = ((size_t)blockIdx.x * blockDim.x + threadIdx.x) * 4;
  int b = (int)(i4 / NTOK);
  int j = (int)(i4 % NTOK);
  const float4 u0 = *(const float4*)(U0 + i4);
  const float4 sm = *(const float4*)(srcmask + i4);
  float uv[4] = {u0.x, u0.y, u0.z, u0.w};
  float sv[4] = {sm.x, sm.y, sm.z, sm.w};
  uint32 res[4];
#pragma unroll
  for (int q = 0; q < 4; ++q) {
    int t = (j + q) / HWREP;
    float cond = (sv[q] > 0.5f) ? -20.7232658f : 0.0f;  // log(1e-9)
    float P = fastlog(uv[q]) + 0.5f * fastlog(Ut[b * TDIM + t]) + cond;
    res[q] = f2key(P);
  }
  *(uint4*)(keys + i4) = make_uint4(res[0], res[1], res[2], res[3]);
}

// ---------------- per-row radix top-K select ----------------
// One workgroup per row; 4 MSB->LSB byte passes. Each pass streams the
// 512 KB key row through two 16 KB LDS tiles via async b128 loads, pipelined:
// issue tile i+1, s_wait_asynccnt 4 (in-order completion => tile i landed),
// barrier, histogram tile i while tile i+1 is in flight.

__device__ __forceinline__ void async_fill(uint32 lds, uint32 goff,
                                           unsigned long long gbase) {
  asm volatile("global_load_async_to_lds_b128 %0, %1, %2 offset:0"
               :: "v"(lds), "v"(goff), "s"(gbase) : "memory");
  asm volatile("global_load_async_to_lds_b128 %0, %1, %2 offset:16"
               :: "v"(lds), "v"(goff), "s"(gbase) : "memory");
  asm volatile("global_load_async_to_lds_b128 %0, %1, %2 offset:32"
               :: "v"(lds), "v"(goff), "s"(gbase) : "memory");
  asm volatile("global_load_async_to_lds_b128 %0, %1, %2 offset:48"
               :: "v"(lds), "v"(goff), "s"(gbase) : "memory");
}

__global__ void k_select(const uint32* __restrict__ keys,
                         const int* __restrict__ Kp,
                         uint32* __restrict__ thresh,
                         uint32* __restrict__ ties,
                         uint32* __restrict__ ctr) {
  __shared__ uint32 tile[2][TILE_U32];
  __shared__ uint32 hist[256];
  __shared__ uint32 s_prefix, s_mask, s_krem;

  const int row = blockIdx.x;
  const int tid = threadIdx.x;
  const unsigned long long gbase =
      (unsigned long long)(uintptr_t)(keys + (size_t)row * NTOK);
  // 32-bit LDS byte addresses of this thread's 64 B staging slots
  const uint32 lds0 = (uint32)(uintptr_t)(&tile[0][0]) + (uint32)tid * 64u;
  const uint32 lds1 = (uint32)(uintptr_t)(&tile[1][0]) + (uint32)tid * 64u;
  const uint32 toff = (uint32)tid * 64u;

  if (tid == 0) { s_prefix = 0u; s_mask = 0u; s_krem = (uint32)Kp[0]; }

  for (int d = 3; d >= 0; --d) {
    hist[tid] = 0u;          // SEL_THREADS == 256 bins
    __syncthreads();         // also publishes s_prefix/s_mask from tid 0
    const uint32 prefix = s_prefix;
    const uint32 pmask  = s_mask;
    const int    shift  = d * 8;

    async_fill(lds0, toff, gbase);               // prologue: tile 0
    for (int it = 0; it < NTILES; ++it) {
      const uint32 cur = (it & 1) ? lds1 : lds0;
      if (it + 1 < NTILES) {
        const uint32 nxt = (it & 1) ? lds0 : lds1;
        async_fill(nxt, toff + (uint32)(it + 1) * TILE_BYTES, gbase);
        // in-order completion: <=4 outstanding => tile `it` has landed
        asm volatile("s_wait_asynccnt 0x4" ::: "memory");
      } else {
        asm volatile("s_wait_asynccnt 0x0" ::: "memory");
      }
      __syncthreads();       // all waves' slices of tile `it` are in LDS
      const uint32* buf = (it & 1) ? &tile[1][0] : &tile[0][0];
#pragma unroll 4
      for (int q = 0; q < 16; ++q) {
        uint32 key = buf[tid * 16 + q];
        if ((key & pmask) == prefix)
          atomicAdd(&hist[(key >> shift) & 0xFFu], 1u);
      }
      __syncthreads();       // tile `it` free for the it+2 async fill
    }

    if (tid == 0) {
      uint32 krem = s_krem, cum = 0u;
      int sel = 0;
      for (int bb = 255; bb >= 0; --bb) {
        uint32 c = hist[bb];
        if (cum + c >= krem) { sel = bb; s_krem = krem - cum; break; }
        cum += c;
      }
      s_prefix = prefix | ((uint32)sel << shift);
      s_mask   = pmask  | (0xFFu << shift);
    }
    __syncthreads();
  }

  if (tid == 0) {
    thresh[row] = s_prefix;  // exact key value of the K-th element
    ties[row]   = s_krem;    // how many == threshold to admit
    ctr[row]    = 0u;        // reset tie counter each call (replay-safe)
  }
}

// ---------------- mask emission (x4 vectorized) ----------------
__global__ void k_mask(const uint32* __restrict__ keys,
                       const uint32* __restrict__ thresh,
                       const uint32* __restrict__ ties,
                       uint32* __restrict__ ctr,
                       float* __restrict__ out) {
  size_t i4 = ((size_t)blockIdx.x * blockDim.x + threadIdx.x) * 4;
  int b = (int)(i4 / NTOK);           // 4 consecutive tokens share a row
  const uint4 k4 = *(const uint4*)(keys + i4);
  const uint32 th = thresh[b];
  uint32 kv[4] = {k4.x, k4.y, k4.z, k4.w};
  float  ov[4];
#pragma unroll
  for (int q = 0; q < 4; ++q) {
    if (kv[q] > th) {
      ov[q] = 1.0f;
    } else if (kv[q] == th) {         // measure-zero for random uniforms
      uint32 p = atomicAdd(&ctr[b], 1u);
      ov[q] = (p < ties[b]) ? 1.0f : 0.0f;
    } else {
      ov[q] = 0.0f;
    }
  }
  *(float4*)(out + i4) = make_float4(ov[0], ov[1], ov[2], ov[3]);
}

extern "C" void kernel_launch(void* const* d_in, const int* in_sizes, int n_in,
                              void* d_out, int out_size, void* d_ws, size_t ws_size,
                              hipStream_t stream) {
  (void)in_sizes; (void)n_in; (void)out_size; (void)ws_size;

  const float* U0_src = (const float*)d_in[0];
  const float* Ut_src = (const float*)d_in[1];
  const float* U0_tgt = (const float*)d_in[2];
  const float* Ut_tgt = (const float*)d_in[3];
  const int*   K_src  = (const int*)d_in[4];
  const int*   K_tgt  = (const int*)d_in[5];

  float* out_src = (float*)d_out;
  float* out_tgt = out_src + (size_t)BATCH * NTOK;

  // workspace: keys (64 MB) + per-row scalars
  uint32* keys   = (uint32*)d_ws;
  uint32* thresh = keys + (size_t)BATCH * NTOK;
  uint32* ties   = thresh + BATCH;
  uint32* ctr    = ties + BATCH;

  const int vec4Blocks = (BATCH * (NTOK / 4)) / 256;  // 16384

  // ---- source mask: top K_src of P_src per row ----
  k_keys_src<<<vec4Blocks, 256, 0, stream>>>(U0_src, Ut_src, keys);
  k_select <<<BATCH, SEL_THREADS, 0, stream>>>(keys, K_src, thresh, ties, ctr);
  k_mask   <<<vec4Blocks, 256, 0, stream>>>(keys, thresh, ties, ctr, out_src);

  // ---- target mask: top K_tgt of P_tgt | ~src ----
  k_keys_tgt<<<vec4Blocks, 256, 0, stream>>>(U0_tgt, Ut_tgt, out_src, keys);
  k_select <<<BATCH, SEL_THREADS, 0, stream>>>(keys, K_tgt, thresh, ties, ctr);
  k_mask   <<<vec4Blocks, 256, 0, stream>>>(keys, thresh, ties, ctr, out_tgt);
}